// WordSpanLoss_21131239096522
// MI455X (gfx1250) — compile-verified
//
#include <hip/hip_runtime.h>
#include <hip/hip_bf16.h>
#include <stdint.h>

// Problem constants (match reference).
#define SPACE_TOKEN 63
#define BB 16
#define FF 512
#define TT 8192
#define WW 4096

typedef __attribute__((ext_vector_type(2))) float v2f;
typedef __attribute__((ext_vector_type(8))) float v8f;

// ---------------------------------------------------------------------------
// CDNA5 async global->LDS path (ASYNCcnt). Guarded so compile never breaks.
// Toolchain prototype (per hipcc diagnostic): int* (AS1) / int* (AS3).
// ---------------------------------------------------------------------------
#ifndef HAVE_ASYNC_LDS
#if defined(__has_builtin)
#if __has_builtin(__builtin_amdgcn_global_load_async_to_lds_b32) && \
    __has_builtin(__builtin_amdgcn_s_wait_asynccnt)
#define HAVE_ASYNC_LDS 1
#endif
#endif
#endif
#ifndef HAVE_ASYNC_LDS
#define HAVE_ASYNC_LDS 0
#endif

#if HAVE_ASYNC_LDS
typedef __attribute__((address_space(1))) int g_int_t;
typedef __attribute__((address_space(3))) int l_int_t;
__device__ __forceinline__ void async_copy_b32(const float* g, float* lds) {
  void* gv = (void*)g;  // drop const
  void* lv = (void*)lds;
  __builtin_amdgcn_global_load_async_to_lds_b32(
      (g_int_t*)gv, (l_int_t*)lv, /*offset=*/0, /*cpol=*/0);
}
__device__ __forceinline__ void wait_async0() {
  __builtin_amdgcn_s_wait_asynccnt(0);
}
#else
__device__ __forceinline__ void async_copy_b32(const float* g, float* lds) {
  *lds = *g;  // synchronous fallback
}
__device__ __forceinline__ void wait_async0() {}
#endif

// ---------------------------------------------------------------------------
// CDNA5 WMMA path for the 16-row time-reduction (all-ones B => row sums).
// ---------------------------------------------------------------------------
#ifndef HAVE_WMMA
#if defined(__has_builtin)
#if __has_builtin(__builtin_amdgcn_wmma_f32_16x16x4_f32)
#define HAVE_WMMA 1
#endif
#endif
#endif
#ifndef HAVE_WMMA
#define HAVE_WMMA 0
#endif

__device__ __forceinline__ float wave_sum32(float v) {
  v += __shfl_xor(v, 16, 32);
  v += __shfl_xor(v, 8, 32);
  v += __shfl_xor(v, 4, 32);
  v += __shfl_xor(v, 2, 32);
  v += __shfl_xor(v, 1, 32);
  return v;
}

// ---------------------------------------------------------------------------
// Kernel 0: zero workspace accumulators (ws is poisoned once, never restored).
// ---------------------------------------------------------------------------
__global__ void ws_init_kernel(unsigned* spanCnt, float* msev, int nslots) {
  int i = blockIdx.x * blockDim.x + threadIdx.x;
  if (i < BB) spanCnt[i] = 0u;
  if (i < nslots) msev[i] = 0.0f;
}

// ---------------------------------------------------------------------------
// Kernel 1: one block per batch. Ordered compaction of space positions, then
// ordered compaction of valid spans into deterministic slots b*capB + rank.
// Valid span k (0-indexed spaces p_0<...<p_{S-1}):
//   k <= S-2, len = p_{k+1}-p_k-1 > 0, word_pos = p_k>>1 (< 4096 always).
// ---------------------------------------------------------------------------
__global__ __launch_bounds__(256) void build_spans_kernel(
    const int* __restrict__ chars, unsigned* __restrict__ spanCnt,
    int4* __restrict__ descs, int capB) {
  __shared__ int s_pos[TT];  // worst case: every char is a space
  __shared__ int s_cnt[256];
  __shared__ int s_total;

  const int b = blockIdx.x;
  const int tid = threadIdx.x;
  const int* tc = chars + (size_t)b * TT + tid * 32;

  // Cache this thread's 32 chars in registers.
  int c[32];
#pragma unroll
  for (int j = 0; j < 32; ++j) c[j] = tc[j];

  // Pass A: count spaces per 32-char chunk.
  int n = 0;
#pragma unroll
  for (int j = 0; j < 32; ++j) n += (c[j] == SPACE_TOKEN);
  s_cnt[tid] = n;
  __syncthreads();
  if (tid == 0) {  // cheap serial exclusive scan over 256 entries
    int acc = 0;
    for (int i = 0; i < 256; ++i) { int v = s_cnt[i]; s_cnt[i] = acc; acc += v; }
    s_total = acc;
  }
  __syncthreads();

  // Pass B: ordered write of space positions.
  int off = s_cnt[tid];
#pragma unroll
  for (int j = 0; j < 32; ++j)
    if (c[j] == SPACE_TOKEN) s_pos[off++] = tid * 32 + j;
  __syncthreads();

  const int S = s_total;
  const int nk = (S > 0) ? (S - 1) : 0;  // candidate spans

  // Pass C: count valid spans per chunk of k, scan, ordered descriptor write.
  int v = 0;
#pragma unroll
  for (int j = 0; j < 32; ++j) {
    int k = tid * 32 + j;
    if (k < nk && (s_pos[k + 1] - s_pos[k]) > 1) v++;
  }
  __syncthreads();
  s_cnt[tid] = v;
  __syncthreads();
  if (tid == 0) {
    int acc = 0;
    for (int i = 0; i < 256; ++i) { int w = s_cnt[i]; s_cnt[i] = acc; acc += w; }
    s_total = acc;
  }
  __syncthreads();
  int rank = s_cnt[tid];
#pragma unroll
  for (int j = 0; j < 32; ++j) {
    int k = tid * 32 + j;
    if (k < nk) {
      int p0 = s_pos[k], p1 = s_pos[k + 1];
      int len = p1 - p0 - 1;
      if (len > 0) {
        if (rank < capB)
          descs[(size_t)b * capB + rank] = make_int4(b, p0 + 1, len, p0 >> 1);
        rank++;
      }
    }
  }
  if (tid == 0) {
    int tot = s_total;
    spanCnt[b] = (unsigned)(tot < capB ? tot : capB);
  }
}

// ---------------------------------------------------------------------------
// Kernel 2: persistent blocks over span slots. Per span:
//   - async-stage wp[:, word_pos] (16KB-stride 4B gather) into LDS (ASYNCcnt),
//   - each of 8 waves reduces 64 feature rows over [t0, t0+len):
//       WMMA path: 4 groups of 16 rows; V_WMMA_F32_16X16X4_F32 with ones-B
//       accumulates 16 running row sums per instruction (EXEC all-ones,
//       uniform control flow, branchless zero-padded tail).
//   - s_wait_asynccnt 0, barrier, squared diffs, block reduce -> mse slot.
// ---------------------------------------------------------------------------
__global__ __launch_bounds__(256) void span_loss_kernel(
    const float* __restrict__ wp, const float* __restrict__ cf,
    const unsigned* __restrict__ spanCnt, const int4* __restrict__ descs,
    int capB, float* __restrict__ msev) {
  __shared__ float s_fsum[FF];
  __shared__ float s_wcol[FF];
  __shared__ float s_wacc[8];

  const int tid = threadIdx.x;
  const int lane = tid & 31;
  const int wv = tid >> 5;
  const int nslots = BB * capB;

  for (int s = blockIdx.x; s < nslots; s += gridDim.x) {
    const int b = s / capB;
    const int rank = s - b * capB;
    if (rank >= (int)spanCnt[b]) continue;  // uniform across the block

    const int4 d = descs[s];
    const int t0 = d.y, len = d.z, wpos = d.w;

    // Stage the scattered word_preds column asynchronously (ASYNCcnt path).
    {
      const float* wcol = wp + ((size_t)b * FF) * WW + wpos;
      async_copy_b32(wcol + (size_t)tid * WW, &s_wcol[tid]);
      async_copy_b32(wcol + (size_t)(tid + 256) * WW, &s_wcol[tid + 256]);
    }

    const float* cfb = cf + ((size_t)b * FF) * TT;

#if HAVE_WMMA
    // Wave wv owns features [wv*64, wv*64+64) as 4 groups of 16 rows.
    // A-fragment (16x4 f32): lanes 0-15 -> M=lane,   K = {0,1}
    //                        lanes 16-31 -> M=lane-16, K = {2,3}
    const int mrow = lane & 15;
    const int koff = (lane >> 4) * 2;
    const v2f ones = {1.0f, 1.0f};
    for (int g = 0; g < 4; ++g) {
      const int fbase = wv * 64 + g * 16;
      const float* rp = cfb + (size_t)(fbase + mrow) * TT + t0;
      if (g + 1 < 4)
        __builtin_prefetch(rp + 16 * TT, 0, 0);  // global_prefetch_b8
      v8f acc = {0.0f, 0.0f, 0.0f, 0.0f, 0.0f, 0.0f, 0.0f, 0.0f};
      for (int c = 0; c < len; c += 4) {  // uniform trip count; EXEC all 1s
        const int ta = c + koff;
        const int tb = ta + 1;
        // Branchless zero-padded tail (clamped in-bounds loads + selects).
        const int ia = (ta < len) ? ta : 0;
        const int ib = (tb < len) ? tb : 0;
        float a0 = rp[ia];
        float a1 = rp[ib];
        a0 = (ta < len) ? a0 : 0.0f;
        a1 = (tb < len) ? a1 : 0.0f;
        const v2f A = {a0, a1};
        acc = __builtin_amdgcn_wmma_f32_16x16x4_f32(
            /*neg_a=*/false, A, /*neg_b=*/false, ones,
            /*c_mod=*/(short)0, acc, /*reuse_a=*/false, /*reuse_b=*/false);
      }
      // D layout: VGPR r, lanes 0-15 -> D[r][lane]; lanes 16-31 -> D[8+r][..].
      // Column 0 of each row holds the row sum; lane 0 has rows 0..7,
      // lane 16 has rows 8..15.
      if (lane == 0) {
#pragma unroll
        for (int r = 0; r < 8; ++r) s_fsum[fbase + r] = acc[r];
      }
      if (lane == 16) {
#pragma unroll
        for (int r = 0; r < 8; ++r) s_fsum[fbase + 8 + r] = acc[r];
      }
    }
#else
    // Fallback: streaming loads + wave32 xor-shuffle reduction per row.
    for (int i = 0; i < 64; ++i) {
      const int f = wv * 64 + i;
      const float* row = cfb + (size_t)f * TT + t0;
      if (i + 1 < 64) __builtin_prefetch(row + TT, 0, 0);
      float sum = 0.0f;
      for (int t = lane; t < len; t += 32) sum += row[t];
      sum = wave_sum32(sum);
      if (lane == 0) s_fsum[f] = sum;
    }
#endif

    wait_async0();    // this wave's async LDS writes are complete
    __syncthreads();  // all waves' s_fsum / s_wcol visible

    const float invLen = 1.0f / (float)len;
    float d0 = s_wcol[tid] - s_fsum[tid] * invLen;
    float d1 = s_wcol[tid + 256] - s_fsum[tid + 256] * invLen;
    float part = wave_sum32(d0 * d0 + d1 * d1);
    if (lane == 0) s_wacc[wv] = part;
    __syncthreads();

    if (tid == 0) {
      float tot = 0.0f;
      for (int i = 0; i < 8; ++i) tot += s_wacc[i];
      msev[s] = tot * (1.0f / (float)FF);  // mean over feature dim
    }
    // s_wcol/s_fsum are only rewritten after every thread passed the barrier
    // above; s_wacc is only rewritten after the next iteration's barrier.
  }
}

// ---------------------------------------------------------------------------
// Kernel 3: deterministic final reduction -> scalar output.
// ---------------------------------------------------------------------------
__global__ __launch_bounds__(256) void finalize_kernel(
    const float* __restrict__ msev, const unsigned* __restrict__ spanCnt,
    int nslots, float* __restrict__ out) {
  __shared__ float red[256];
  const int tid = threadIdx.x;
  float acc = 0.0f;
  for (int i = tid; i < nslots; i += 256) acc += msev[i];
  red[tid] = acc;
  __syncthreads();
  for (int stride = 128; stride > 0; stride >>= 1) {
    if (tid < stride) red[tid] += red[tid + stride];
    __syncthreads();
  }
  if (tid == 0) {
    unsigned cnt = 0;
    for (int b = 0; b < BB; ++b) cnt += spanCnt[b];
    out[0] = (cnt > 0u) ? (red[0] / (float)cnt) : 0.0f;
  }
}

// ---------------------------------------------------------------------------
// Host launcher.
// ---------------------------------------------------------------------------
extern "C" void kernel_launch(void* const* d_in, const int* in_sizes, int n_in,
                              void* d_out, int out_size, void* d_ws,
                              size_t ws_size, hipStream_t stream) {
  const float* wp = (const float*)d_in[0];  // (B, F, W) f32
  const float* cf = (const float*)d_in[1];  // (B, F, T) f32
  const int* ch = (const int*)d_in[2];      // (B, T)    i32
  float* out = (float*)d_out;               // scalar f32

  // Workspace layout: [16B pad][spanCnt: 16 u32 @16][pad to 128]
  //                   [descs: int4 x BB*capB @128][msev: f32 x BB*capB]
  size_t avail = (ws_size > 128) ? (ws_size - 128) : 0;
  size_t capB_sz = avail / ((size_t)BB * 20);
  int capB = (capB_sz > 4096) ? 4096 : (int)capB_sz;
  if (capB < 1) capB = 1;
  const int nslots = BB * capB;

  unsigned* spanCnt = (unsigned*)((char*)d_ws + 16);
  int4* descs = (int4*)((char*)d_ws + 128);
  float* msev = (float*)((char*)d_ws + 128 + (size_t)nslots * sizeof(int4));

  ws_init_kernel<<<(nslots + 255) / 256, 256, 0, stream>>>(spanCnt, msev,
                                                           nslots);
  build_spans_kernel<<<BB, 256, 0, stream>>>(ch, spanCnt, descs, capB);
  span_loss_kernel<<<2048, 256, 0, stream>>>(wp, cf, spanCnt, descs, capB,
                                             msev);
  finalize_kernel<<<1, 256, 0, stream>>>(msev, spanCnt, nslots, out);
}